// CrossDMHAttention_5677946765401
// MI455X (gfx1250) — compile-verified
//
#include <hip/hip_runtime.h>

typedef float v2f __attribute__((ext_vector_type(2)));
typedef float v4f __attribute__((ext_vector_type(4)));
typedef float v8f __attribute__((ext_vector_type(8)));

#define DQ     13
#define DKV    32
#define T_CTX  24
#define A_DIM  32
#define HD     8
#define O_DIM  32
#define WAVES  4
#define BPW    16            // batch rows per wave (WMMA M)
#define BPB    (WAVES*BPW)   // 64 batch rows per block
#define SCALE  0.35355339059327373f  // 1/sqrt(8)

// ---- LDS layout (floats) ----
// Weights stored pre-swizzled in B-operand layout:
//   swz[(c*2+n)*64 + lane*2 + r] = W[4c + 2*(lane>>4) + r][16n + (lane&15)]
// so each B operand is ONE aligned ds_load_b64 into an adjacent VGPR pair.
#define OFF_WQ   0        // 4 chunks  (K=13 padded to 16) -> 512
#define OFF_WK   512      // 8 chunks -> 1024
#define OFF_WV   1536
#define OFF_WO   2560
#define OFF_WD1  3584
#define OFF_WD2  4608
#define OFF_LNW  5632
#define OFF_LNB  5664
#define OFF_WAVE 5696
// per wave: kv0(512) | kv1(512) | K/bufB(512) | V/bufC(512)
#define WAVE_STRIDE 2048
#define SMEM_FLOATS (OFF_WAVE + WAVES*WAVE_STRIDE)  // 13888 floats = 55.5 KB

static __device__ __forceinline__ v8f wmma4(v2f a, v2f b, v8f c) {
  // V_WMMA_F32_16X16X4_F32 : D = A(16x4) x B(4x16) + C(16x16), all f32
  return __builtin_amdgcn_wmma_f32_16x16x4_f32(
      false, a, false, b, (short)0, c, false, false);
}

static __device__ __forceinline__ float gelu_exact(float x) {
  return 0.5f * x * (1.0f + erff(x * 0.70710678118654752f));
}

// scheduling-only barrier: intra-wave cross-lane LDS ordering is guaranteed by
// in-order same-wave LDS execution; this just pins compiler ordering.
static __device__ __forceinline__ void wbar() { __builtin_amdgcn_wave_barrier(); }

// fetch pre-swizzled B operand: one ds_load_b64
static __device__ __forceinline__ v2f bop(const float* swz, int c, int n, int lane) {
  return *(const v2f*)&swz[(c * 2 + n) * 64 + lane * 2];
}

__global__ __launch_bounds__(WAVES * 32)
void cross_dmha_kernel(const float* __restrict__ query,
                       const float* __restrict__ kv,
                       const float* __restrict__ Wq,
                       const float* __restrict__ Wk,
                       const float* __restrict__ Wv,
                       const float* __restrict__ Wo,
                       const float* __restrict__ lnw,
                       const float* __restrict__ lnb,
                       const float* __restrict__ Wd1,
                       const float* __restrict__ Wd2,
                       float* __restrict__ out) {
  __shared__ float smem[SMEM_FLOATS];

  const int tid  = threadIdx.x;
  const int lane = tid & 31;
  const int w    = tid >> 5;
  const int bb   = lane & 15;        // M row within tile (batch)
  const int hl   = lane >> 4;        // lane half: K/N split + head-pair id
  const int batchBase = blockIdx.x * BPB + w * BPW;

  // ---- stage weights into LDS, pre-swizzled to B-operand layout ----
  for (int i = tid; i < 512; i += WAVES * 32) {   // Wq: 4 chunks, K padded
    int c = i >> 7, n = (i >> 6) & 1, ln = (i >> 1) & 31, r = i & 1;
    int kk = 4 * c + 2 * (ln >> 4) + r;
    int col = 16 * n + (ln & 15);
    smem[OFF_WQ + i] = (kk < DQ) ? Wq[kk * A_DIM + col] : 0.0f;
  }
  for (int i = tid; i < 1024; i += WAVES * 32) {  // 8-chunk weights
    int c = i >> 7, n = (i >> 6) & 1, ln = (i >> 1) & 31, r = i & 1;
    int idx = (4 * c + 2 * (ln >> 4) + r) * 32 + 16 * n + (ln & 15);
    smem[OFF_WK  + i] = Wk[idx];
    smem[OFF_WV  + i] = Wv[idx];
    smem[OFF_WO  + i] = Wo[idx];
    smem[OFF_WD1 + i] = Wd1[idx];
    smem[OFF_WD2 + i] = Wd2[idx];
  }
  if (tid < 32) { smem[OFF_LNW + tid] = lnw[tid]; smem[OFF_LNB + tid] = lnb[tid]; }
  __syncthreads();   // the only cross-wave sync

  float* bufKV0 = &smem[OFF_WAVE + w * WAVE_STRIDE];  // kv tile double buffer
  float* bufB   = bufKV0 + 1024;                      // K tile / scratch
  float* bufC   = bufKV0 + 1536;                      // V tile / scratch
  float* postA  = bufKV0;                             // ctx rows after loop
  float* bufLN  = bufKV0 + 512;                       // ln_out rows after loop

  // ---- per-lane async-copy geometry for kv tiles (16 rows x 128B) ----
  const unsigned long long kvbase = (unsigned long long)(size_t)kv;
  const unsigned ldsWaveB = (unsigned)(size_t)bufKV0;   // LDS byte offset of kv0
  unsigned ldsoff[4], goff[4];
#pragma unroll
  for (int k = 0; k < 4; ++k) {
    int idx = k * 32 + lane;                 // 0..127
    int row = idx >> 3, c4 = idx & 7;
    ldsoff[k] = (unsigned)(row * 128 + c4 * 16);
    goff[k]   = (unsigned)(batchBase + row) * (T_CTX * DKV * 4) + (unsigned)(c4 * 16);
  }

  // prologue: async-load kv tile t=0 into kv0 (ASYNCcnt += 4), NT streaming
#pragma unroll
  for (int k = 0; k < 4; ++k) {
    unsigned la = ldsWaveB + ldsoff[k];
    unsigned ga = goff[k];
    asm volatile("global_load_async_to_lds_b128 %0, %1, %2 th:TH_LOAD_NT"
                 :: "v"(la), "v"(ga), "s"(kvbase) : "memory");
  }

  // =========== Q projection: [16 x 13(pad16)] @ Wq[13(pad16) x 32] ===========
  for (int i = lane; i < 256; i += 32) {              // stage query -> bufB [16][16]
    int r = i >> 4, c = i & 15;
    bufB[i] = (c < DQ) ? query[(size_t)(batchBase + r) * DQ + c] : 0.0f;
  }
  wbar();

  v8f Qd0 = {}, Qd1 = {};
#pragma unroll
  for (int c = 0; c < 4; ++c) {               // K chunks of 4 (K = 16 padded)
    v2f a = *(const v2f*)&bufB[bb * 16 + 4 * c + 2 * hl];
    Qd0 = wmma4(a, bop(&smem[OFF_WQ], c, 0, lane), Qd0);
    Qd1 = wmma4(a, bop(&smem[OFF_WQ], c, 1, lane), Qd1);
  }
  wbar();
#pragma unroll
  for (int r = 0; r < 8; ++r) {               // D -> bufC rows [m][ch]
    bufC[(r + 8 * hl) * 32 + bb]      = Qd0[r];
    bufC[(r + 8 * hl) * 32 + 16 + bb] = Qd1[r];
  }
  wbar();
  float qv[16];                               // lane owns batch bb, heads 2hl,2hl+1
#pragma unroll
  for (int j4 = 0; j4 < 4; ++j4) {
    v4f q4 = *(const v4f*)&bufC[bb * 32 + 16 * hl + 4 * j4];
#pragma unroll
    for (int e = 0; e < 4; ++e) qv[4 * j4 + e] = q4[e];
  }
  wbar();

  // ---- preload Wk/Wv B-operands into VGPRs (loop-invariant, ds_load_b64 each) ----
  v2f wkB[8][2], wvB[8][2];
#pragma unroll
  for (int c = 0; c < 8; ++c)
#pragma unroll
    for (int n = 0; n < 2; ++n) {
      wkB[c][n] = bop(&smem[OFF_WK], c, n, lane);
      wvB[c][n] = bop(&smem[OFF_WV], c, n, lane);
    }

  // ---- online-softmax attention over T, K/V projected on the fly ----
  float m0 = -1e30f, m1 = -1e30f, l0 = 0.0f, l1 = 0.0f;
  float ctx[16];
#pragma unroll
  for (int j = 0; j < 16; ++j) ctx[j] = 0.0f;

#pragma unroll 1
  for (int t = 0; t < T_CTX; ++t) {
    // issue async copy of tile t+1 into the other buffer, then wait for tile t
    if (t + 1 < T_CTX) {
      unsigned dstB = ldsWaveB + (unsigned)(((t + 1) & 1) * 2048);
      unsigned tadd = (unsigned)(t + 1) * (DKV * 4);
#pragma unroll
      for (int k = 0; k < 4; ++k) {
        unsigned la = dstB + ldsoff[k];
        unsigned ga = goff[k] + tadd;
        asm volatile("global_load_async_to_lds_b128 %0, %1, %2 th:TH_LOAD_NT"
                     :: "v"(la), "v"(ga), "s"(kvbase) : "memory");
      }
      // in-order completion: <=4 outstanding means tile t is fully in LDS
      asm volatile("s_wait_asynccnt 0x4" ::: "memory");
    } else {
      asm volatile("s_wait_asynccnt 0x0" ::: "memory");
    }

    const float* kvcur = bufKV0 + ((t & 1) ? 512 : 0);

    // K/V projection for this t: 8 K-chunks x {K,V} x {N0,N1}
    v8f Kd0 = {}, Kd1 = {}, Vd0 = {}, Vd1 = {};
#pragma unroll
    for (int c = 0; c < 8; ++c) {
      v2f a = *(const v2f*)&kvcur[bb * 32 + 4 * c + 2 * hl];
      Kd0 = wmma4(a, wkB[c][0], Kd0);
      Kd1 = wmma4(a, wkB[c][1], Kd1);
      Vd0 = wmma4(a, wvB[c][0], Vd0);
      Vd1 = wmma4(a, wvB[c][1], Vd1);
    }
    // D -> LDS rows
#pragma unroll
    for (int r = 0; r < 8; ++r) {
      int row = r + 8 * hl;
      bufB[row * 32 + bb]      = Kd0[r];
      bufB[row * 32 + 16 + bb] = Kd1[r];
      bufC[row * 32 + bb]      = Vd0[r];
      bufC[row * 32 + 16 + bb] = Vd1[r];
    }
    wbar();

    // per-lane flash-softmax update for 2 heads
    float kkv[16], vvv[16];
#pragma unroll
    for (int j4 = 0; j4 < 4; ++j4) {
      v4f k4 = *(const v4f*)&bufB[bb * 32 + 16 * hl + 4 * j4];
      v4f v4 = *(const v4f*)&bufC[bb * 32 + 16 * hl + 4 * j4];
#pragma unroll
      for (int e = 0; e < 4; ++e) { kkv[4 * j4 + e] = k4[e]; vvv[4 * j4 + e] = v4[e]; }
    }
    float s0 = 0.0f, s1 = 0.0f;
#pragma unroll
    for (int j = 0; j < 8; ++j) { s0 += qv[j] * kkv[j]; s1 += qv[j + 8] * kkv[j + 8]; }
    s0 *= SCALE; s1 *= SCALE;

    float m0n = fmaxf(m0, s0);
    float a0  = __expf(m0 - m0n);
    float p0  = __expf(s0 - m0n);
    l0 = l0 * a0 + p0; m0 = m0n;
#pragma unroll
    for (int j = 0; j < 8; ++j) ctx[j] = ctx[j] * a0 + p0 * vvv[j];

    float m1n = fmaxf(m1, s1);
    float a1  = __expf(m1 - m1n);
    float p1  = __expf(s1 - m1n);
    l1 = l1 * a1 + p1; m1 = m1n;
#pragma unroll
    for (int j = 0; j < 8; ++j) ctx[j + 8] = ctx[j + 8] * a1 + p1 * vvv[j + 8];
    wbar();
  }

  // normalize and write ctx rows -> postA[m][ch]
  {
    float il0 = 1.0f / l0, il1 = 1.0f / l1;
#pragma unroll
    for (int j = 0; j < 8; ++j) {
      postA[bb * 32 + 16 * hl + j]     = ctx[j] * il0;
      postA[bb * 32 + 16 * hl + 8 + j] = ctx[j + 8] * il1;
    }
  }
  wbar();

  // =========== out = ctx @ Wo ===========
  v8f Od0 = {}, Od1 = {};
#pragma unroll
  for (int c = 0; c < 8; ++c) {
    v2f a = *(const v2f*)&postA[bb * 32 + 4 * c + 2 * hl];
    Od0 = wmma4(a, bop(&smem[OFF_WO], c, 0, lane), Od0);
    Od1 = wmma4(a, bop(&smem[OFF_WO], c, 1, lane), Od1);
  }
#pragma unroll
  for (int r = 0; r < 8; ++r) {
    bufB[(r + 8 * hl) * 32 + bb]      = Od0[r];
    bufB[(r + 8 * hl) * 32 + 16 + bb] = Od1[r];
  }
  wbar();

  // =========== LayerNorm per row (lanes 0..15 own one row each) ===========
  if (hl == 0) {
    float sum = 0.0f, sq = 0.0f;
#pragma unroll
    for (int i4 = 0; i4 < 8; ++i4) {
      v4f x4 = *(const v4f*)&bufB[bb * 32 + 4 * i4];
#pragma unroll
      for (int e = 0; e < 4; ++e) { sum += x4[e]; sq += x4[e] * x4[e]; }
    }
    float mu  = sum * (1.0f / 32.0f);
    float var = sq * (1.0f / 32.0f) - mu * mu;
    float rs  = rsqrtf(var + 1e-5f);
#pragma unroll
    for (int i = 0; i < 32; ++i) {
      float x = bufB[bb * 32 + i];
      bufLN[bb * 32 + i] = (x - mu) * rs * smem[OFF_LNW + i] + smem[OFF_LNB + i];
    }
  }
  wbar();

  // =========== MLP1: h = gelu(ln_out @ Wd1) ===========
  v8f Hd0 = {}, Hd1 = {};
#pragma unroll
  for (int c = 0; c < 8; ++c) {
    v2f a = *(const v2f*)&bufLN[bb * 32 + 4 * c + 2 * hl];
    Hd0 = wmma4(a, bop(&smem[OFF_WD1], c, 0, lane), Hd0);
    Hd1 = wmma4(a, bop(&smem[OFF_WD1], c, 1, lane), Hd1);
  }
#pragma unroll
  for (int r = 0; r < 8; ++r) {
    bufC[(r + 8 * hl) * 32 + bb]      = gelu_exact(Hd0[r]);
    bufC[(r + 8 * hl) * 32 + 16 + bb] = gelu_exact(Hd1[r]);
  }
  wbar();

  // =========== MLP2: h = gelu(h @ Wd2); final = ln_out + h ===========
  v8f Gd0 = {}, Gd1 = {};
#pragma unroll
  for (int c = 0; c < 8; ++c) {
    v2f a = *(const v2f*)&bufC[bb * 32 + 4 * c + 2 * hl];
    Gd0 = wmma4(a, bop(&smem[OFF_WD2], c, 0, lane), Gd0);
    Gd1 = wmma4(a, bop(&smem[OFF_WD2], c, 1, lane), Gd1);
  }
#pragma unroll
  for (int r = 0; r < 8; ++r) {
    int m = r + 8 * hl;
    float f0 = bufLN[m * 32 + bb]      + gelu_exact(Gd0[r]);
    float f1 = bufLN[m * 32 + 16 + bb] + gelu_exact(Gd1[r]);
    out[(size_t)(batchBase + m) * O_DIM + bb]      = f0;
    out[(size_t)(batchBase + m) * O_DIM + 16 + bb] = f1;
  }
}

extern "C" void kernel_launch(void* const* d_in, const int* in_sizes, int n_in,
                              void* d_out, int out_size, void* d_ws, size_t ws_size,
                              hipStream_t stream) {
  (void)n_in; (void)out_size; (void)d_ws; (void)ws_size;
  const float* query = (const float*)d_in[0];
  const float* kv    = (const float*)d_in[1];
  const float* Wq    = (const float*)d_in[2];
  const float* Wk    = (const float*)d_in[3];
  const float* Wv    = (const float*)d_in[4];
  const float* Wo    = (const float*)d_in[5];
  const float* lnw   = (const float*)d_in[6];
  const float* lnb   = (const float*)d_in[7];
  const float* Wd1   = (const float*)d_in[8];
  const float* Wd2   = (const float*)d_in[9];
  float* out = (float*)d_out;

  int B = in_sizes[0] / DQ;          // 131072
  dim3 grid(B / BPB), block(WAVES * 32);
  cross_dmha_kernel<<<grid, block, 0, stream>>>(query, kv, Wq, Wk, Wv, Wo,
                                                lnw, lnb, Wd1, Wd2, out);
}